// TextRNNRegression_74062416052722
// MI455X (gfx1250) — compile-verified
//
#include <hip/hip_runtime.h>
#include <math.h>

// MI455X (gfx1250) fused TextRNN forward.
// ~8.6 GFLOP fp32; all operands L2/LDS/VGPR-resident -> latency-bound on the
// T=512 sequential scan. One kernel: 32 WGs (one 16-row batch strip each),
// 8 wave32s, two-layer software pipeline (layer-2 runs one step behind
// layer-1), fp32 WMMA, register-resident weights, cross-iteration prefetch of
// the embedding operand so no global-load latency sits on the recurrence path.

typedef __attribute__((ext_vector_type(2))) float v2f;
typedef __attribute__((ext_vector_type(8))) float v8f;

#define EMB 64
#define HID 64
#define BATCH 512
#define SEQT 512
#define LDSTRIDE 66   // pad 16x64 fp32 tiles: bank = (2*row + col) % 64 -> conflict-free

#if __has_builtin(__builtin_amdgcn_tanhf)
#define FAST_TANH(v) __builtin_amdgcn_tanhf(v)
#elif __has_builtin(__builtin_amdgcn_tanh_f32)
#define FAST_TANH(v) __builtin_amdgcn_tanh_f32(v)
#else
#define FAST_TANH(v) tanhf(v)
#endif

__device__ __forceinline__ v8f wmma_f32(v2f a, v2f b, v8f c) {
  // V_WMMA_F32_16X16X4_F32: D = A(16x4 f32) * B(4x16 f32) + C(16x16 f32)
  return __builtin_amdgcn_wmma_f32_16x16x4_f32(
      /*neg_a=*/false, a, /*neg_b=*/false, b,
      /*c_mod=*/(short)0, c, /*reuse_a=*/false, /*reuse_b=*/false);
}

__global__ __launch_bounds__(256, 1)
void rnn_fused_kernel(const int* __restrict__ x,
                      const float* __restrict__ emb,
                      const float* __restrict__ W_ih0,
                      const float* __restrict__ W_hh0,
                      const float* __restrict__ b_ih0,
                      const float* __restrict__ b_hh0,
                      const float* __restrict__ W_ih1,
                      const float* __restrict__ W_hh1,
                      const float* __restrict__ b_ih1,
                      const float* __restrict__ b_hh1,
                      const float* __restrict__ fc_w,
                      const float* __restrict__ fc_b,
                      float* __restrict__ out)
{
  __shared__ float h1buf[2][16 * LDSTRIDE];  // double-buffered layer-1 state
  __shared__ float h2buf[2][16 * LDSTRIDE];  // double-buffered layer-2 state

  const int tid   = threadIdx.x;
  const int lane  = tid & 31;
  const int wave  = tid >> 5;
  const int half  = lane >> 4;     // K-half selector for A/B fragments
  const int lrow  = lane & 15;     // M row (A) / N col (B,C,D) within tile
  const int strip = blockIdx.x;    // 16 batch rows per workgroup
  const bool isL1 = (wave < 4);    // waves 0-3: layer-1 chain; 4-7: layer-2 chain
  const int nt    = isL1 ? wave : (wave - 4);  // 16-wide output column tile
  const int col   = nt * 16 + lrow;            // output column this lane owns

  // h[-1] = 0 for both layers, both parities.
  for (int idx = tid; idx < 16 * LDSTRIDE; idx += 256) {
    h1buf[0][idx] = 0.f; h1buf[1][idx] = 0.f;
    h2buf[0][idx] = 0.f; h2buf[1][idx] = 0.f;
  }

  // Register-resident B fragments. out = h @ W^T => B[k][n] = W[n][k]; the
  // per-lane K=4 fragment is a contiguous float2 of row-major W:
  //   lanes 0-15: K=k0..k0+1 at N=lane; lanes 16-31: K=k0+2..k0+3 at N=lane-16.
  v2f wbA[16], wbB[16];
  {
    const float* WA = isL1 ? W_ih0 : W_ih1;   // input-side weight
    const float* WB = isL1 ? W_hh0 : W_hh1;   // hidden-side weight
    const int base = col * HID + 2 * half;
#pragma unroll
    for (int k = 0; k < 16; ++k) {
      wbA[k] = *(const v2f*)(WA + base + 4 * k);
      wbB[k] = *(const v2f*)(WB + base + 4 * k);
    }
  }

  const float bias = isL1 ? (b_ih0[col] + b_hh0[col]) : (b_ih1[col] + b_hh1[col]);
  const int xrow = (strip * 16 + lrow) * SEQT;  // this lane's token stream

  // L1 prologue: prefetch step-0 embedding A-fragments and step-1 token, so the
  // steady-state loop never has a global load on its critical path.
  v2f eA[16];
  int tok_next = 0;
  if (isL1) {
    const int tok0 = x[xrow + 0];
    const float* erow = emb + tok0 * EMB + 2 * half;
#pragma unroll
    for (int k = 0; k < 16; ++k) eA[k] = *(const v2f*)(erow + 4 * k);
    tok_next = x[xrow + 1];
  }

  __syncthreads();

  // Iteration i: layer-1 computes h1[i] while layer-2 computes h2[i-1].
  // Both consume only state produced at iteration i-1 (one barrier per iter).
  for (int i = 0; i <= SEQT; ++i) {
    if (isL1) {
      if (i < SEQT) {
        // h1[i] = tanh(h1[i-1] @ W_hh0^T + e(x[:,i]) @ W_ih0^T + b_ih0 + b_hh0)
        v8f acc;
#pragma unroll
        for (int r = 0; r < 8; ++r) acc[r] = bias;
        // Recurrent GEMM first: LDS operands are ready right after the barrier.
        const float* h1p = &h1buf[(i + 1) & 1][lrow * LDSTRIDE + 2 * half]; // h1[i-1]
#pragma unroll
        for (int k = 0; k < 16; ++k) {
          v2f a = *(const v2f*)(h1p + 4 * k);
          acc = wmma_f32(a, wbB[k], acc);
        }
        // Input GEMM with fragments prefetched during iteration i-1.
#pragma unroll
        for (int k = 0; k < 16; ++k)
          acc = wmma_f32(eA[k], wbA[k], acc);
        // Prefetch step i+1 fragments (token was prefetched last iteration);
        // these loads have the barrier + next recurrent GEMM to land.
        {
          const float* erow = emb + tok_next * EMB + 2 * half;
#pragma unroll
          for (int k = 0; k < 16; ++k) eA[k] = *(const v2f*)(erow + 4 * k);
          const int inext = (i + 2 < SEQT) ? (i + 2) : (SEQT - 1);
          tok_next = x[xrow + inext];
        }
        float* dst = &h1buf[i & 1][(8 * half) * LDSTRIDE + col];
#pragma unroll
        for (int r = 0; r < 8; ++r) dst[r * LDSTRIDE] = FAST_TANH(acc[r]);
      }
    } else {
      if (i >= 1) {
        // j = i-1: h2[j] = tanh(h1[j] @ W_ih1^T + h2[j-1] @ W_hh1^T + biases)
        v8f acc;
#pragma unroll
        for (int r = 0; r < 8; ++r) acc[r] = bias;
        const float* h1c = &h1buf[(i + 1) & 1][lrow * LDSTRIDE + 2 * half]; // h1[j]
        const float* h2p = &h2buf[i & 1][lrow * LDSTRIDE + 2 * half];       // h2[j-1]
#pragma unroll
        for (int k = 0; k < 16; ++k) {
          v2f a = *(const v2f*)(h1c + 4 * k);
          acc = wmma_f32(a, wbA[k], acc);
        }
#pragma unroll
        for (int k = 0; k < 16; ++k) {
          v2f a = *(const v2f*)(h2p + 4 * k);
          acc = wmma_f32(a, wbB[k], acc);
        }
        float* dst = &h2buf[(i + 1) & 1][(8 * half) * LDSTRIDE + col];      // h2[j]
#pragma unroll
        for (int r = 0; r < 8; ++r) dst[r * LDSTRIDE] = FAST_TANH(acc[r]);
      }
    }
    __syncthreads();
  }

  // Final FC: h2[T-1] sits in parity (T-1)&1 == 1. out[b] = h2[b] . fc_w + fc_b
  if (tid < 16) {
    float s = fc_b[0];
    const float* row = &h2buf[(SEQT - 1) & 1][tid * LDSTRIDE];
#pragma unroll
    for (int h = 0; h < HID; ++h) s += row[h] * fc_w[h];
    out[strip * 16 + tid] = s;
  }
}

extern "C" void kernel_launch(void* const* d_in, const int* in_sizes, int n_in,
                              void* d_out, int out_size, void* d_ws, size_t ws_size,
                              hipStream_t stream) {
  (void)in_sizes; (void)n_in; (void)d_ws; (void)ws_size; (void)out_size;
  const int*   x     = (const int*)  d_in[0];
  const float* emb   = (const float*)d_in[1];
  const float* W_ih0 = (const float*)d_in[2];
  const float* W_hh0 = (const float*)d_in[3];
  const float* b_ih0 = (const float*)d_in[4];
  const float* b_hh0 = (const float*)d_in[5];
  const float* W_ih1 = (const float*)d_in[6];
  const float* W_hh1 = (const float*)d_in[7];
  const float* b_ih1 = (const float*)d_in[8];
  const float* b_hh1 = (const float*)d_in[9];
  const float* fc_w  = (const float*)d_in[10];
  const float* fc_b  = (const float*)d_in[11];

  rnn_fused_kernel<<<dim3(BATCH / 16), dim3(256), 0, stream>>>(
      x, emb, W_ih0, W_hh0, b_ih0, b_hh0,
      W_ih1, W_hh1, b_ih1, b_hh1, fc_w, fc_b, (float*)d_out);
}